// DecisionTreeModule_57999238365558
// MI455X (gfx1250) — compile-verified
//
#include <hip/hip_runtime.h>
#include <hip/hip_bf16.h>
#include <math.h>

// ---------------------------------------------------------------------------
// DecisionTree forward on MI455X (gfx1250).
//
// Softmax over leaf_probabilities depends only on the leaf index (4096
// leaves), so we precompute softmax once into a 1.6MB table (L2-resident) and
// per-sample work collapses to a 12-level traversal + 400B row copy.
// Memory-bound: ~0.7 GB total traffic -> ~30us floor at 23.3 TB/s.
// No matmul content -> no WMMA; CDNA5 paths used: wave32 shuffles, async
// global->LDS staging (ASYNCcnt / s_wait_asynccnt), global_prefetch_b8 row
// prefetch, NT streaming stores.
// ---------------------------------------------------------------------------

#define INPUT_DIM   256
#define N_CLASSES   100
#define N_CLS4      25     // 100 floats = 25 x float4
#define MAX_DEPTH   12
#define N_NODES     4095   // 2^12 - 1
#define N_LEAVES    4096
#define BLK         256

typedef float vf4 __attribute__((ext_vector_type(4)));

// ---- async global->LDS staging (gfx1250), guarded so both toolchains build --
#if defined(__has_builtin)
#if __has_builtin(__builtin_amdgcn_global_load_async_to_lds_b128) && \
    __has_builtin(__builtin_amdgcn_global_load_async_to_lds_b32)
#define USE_ASYNC_LDS 1
#endif
#endif

#ifdef USE_ASYNC_LDS
// Builtin prototypes (confirmed by clang diagnostics in earlier round):
//   b128: (v4i addrspace(1)*, v4i addrspace(3)*, imm int offset, imm int cpol)
//   b32 : (int addrspace(1)*, int addrspace(3)*, imm int offset, imm int cpol)
typedef int v4i __attribute__((vector_size(4 * sizeof(int))));
#define GAS128(p) ((__attribute__((address_space(1))) v4i*)(p))
#define LAS128(p) ((__attribute__((address_space(3))) v4i*)(p))
#define GAS32(p)  ((__attribute__((address_space(1))) int*)(p))
#define LAS32(p)  ((__attribute__((address_space(3))) int*)(p))

__device__ __forceinline__ void async_wait0() {
#if __has_builtin(__builtin_amdgcn_s_wait_asynccnt)
  __builtin_amdgcn_s_wait_asynccnt(0);
#else
  asm volatile("s_wait_asynccnt 0" ::: "memory");
#endif
}
#endif

// ---------------------------------------------------------------------------
// Kernel 1: softmax table over the 4096 leaf rows. One wave32 per row.
// ---------------------------------------------------------------------------
__global__ __launch_bounds__(BLK)
void dt_softmax_table_kernel(const float* __restrict__ lp,
                             float* __restrict__ table) {
  const int wave = threadIdx.x >> 5;
  const int lane = threadIdx.x & 31;
  const int row  = blockIdx.x * (BLK >> 5) + wave;
  if (row >= N_LEAVES) return;

  const float* src = lp + (size_t)row * N_CLASSES;

  float v[4];
  float m = -INFINITY;
#pragma unroll
  for (int k = 0; k < 4; ++k) {
    const int j = lane + 32 * k;
    v[k] = (j < N_CLASSES) ? src[j] : -INFINITY;
    m = fmaxf(m, v[k]);
  }
#pragma unroll
  for (int off = 16; off >= 1; off >>= 1)   // wave32 butterfly max
    m = fmaxf(m, __shfl_xor(m, off, 32));

  float s = 0.0f;
#pragma unroll
  for (int k = 0; k < 4; ++k) {
    const int j = lane + 32 * k;
    v[k] = (j < N_CLASSES) ? expf(v[k] - m) : 0.0f;
    s += v[k];
  }
#pragma unroll
  for (int off = 16; off >= 1; off >>= 1)   // wave32 butterfly sum
    s += __shfl_xor(s, off, 32);

  const float inv = 1.0f / s;
#pragma unroll
  for (int k = 0; k < 4; ++k) {
    const int j = lane + 32 * k;
    if (j < N_CLASSES) table[(size_t)row * N_CLASSES + j] = v[k] * inv;
  }
}

// ---------------------------------------------------------------------------
// Kernel 2: tree traversal + gather of precomputed softmax rows.
//   Phase 0: async-stage the 32KB tree (features+thresholds) into LDS.
//   Phase 1: one thread per sample; prefetch the whole 1KB x-row (8 cache
//            lines, independent global_prefetch_b8), then 12 LDS node lookups
//            + 12 near-cache gathers (no serial HBM latency chain).
//   Phase 2: coalesced cooperative copy, B128 granularity, NT stores.
// ---------------------------------------------------------------------------
__global__ __launch_bounds__(BLK)
void dt_traverse_kernel(const float* __restrict__ x,
                        const float* __restrict__ sf,
                        const float* __restrict__ st,
                        const float* __restrict__ table,
                        float* __restrict__ out,
                        int batch) {
  __shared__ __align__(16) float thrL[N_NODES];
  __shared__ __align__(16) int   featL[N_NODES];   // staged as float bits
  __shared__ int leafL[BLK];

  const int tid = threadIdx.x;

  // ---- Phase 0: stage tree into LDS ----------------------------------------
#ifdef USE_ASYNC_LDS
  // 4095 floats = 1023 x b128 + 3 x b32 per array. ASYNCcnt-tracked DMA;
  // no VGPR data movement.
  for (int c = tid; c < 1023; c += BLK) {
    __builtin_amdgcn_global_load_async_to_lds_b128(GAS128(st + 4 * c),
                                                   LAS128(&thrL[4 * c]), 0, 0);
    __builtin_amdgcn_global_load_async_to_lds_b128(GAS128(sf + 4 * c),
                                                   LAS128(&featL[4 * c]), 0, 0);
  }
  if (tid < 3) {
    __builtin_amdgcn_global_load_async_to_lds_b32(GAS32(st + 4092 + tid),
                                                  LAS32(&thrL[4092 + tid]), 0, 0);
    __builtin_amdgcn_global_load_async_to_lds_b32(GAS32(sf + 4092 + tid),
                                                  LAS32(&featL[4092 + tid]), 0, 0);
  }
  async_wait0();
#else
  for (int i = tid; i < N_NODES; i += BLK) {
    thrL[i]  = st[i];
    featL[i] = __float_as_int(sf[i]);
  }
#endif
  __syncthreads();

  // Convert split_features (float) -> clamped int feature index, once per node.
  for (int i = tid; i < N_NODES; i += BLK) {
    const float f = __int_as_float(featL[i]);
    int fi = (int)floorf(f);
    fi = fi < 0 ? 0 : (fi > INPUT_DIM - 1 ? INPUT_DIM - 1 : fi);
    featL[i] = fi;
  }
  __syncthreads();

  // ---- Phase 1: traversal, one sample per thread ---------------------------
  const int s0 = blockIdx.x * BLK;
  const int b  = s0 + tid;
  int leaf = 0;
  if (b < batch) {
    const float* xr = x + (size_t)b * INPUT_DIM;
    // Prefetch all 8 x 128B lines of this sample's row (independent requests)
    // so the 12 dependent gathers below hit near cache instead of forming a
    // 12-deep HBM latency chain. Also turns x into a sequential 1KB stream.
#pragma unroll
    for (int c = 0; c < INPUT_DIM; c += 32)          // 32 floats = 128B line
      __builtin_prefetch(xr + c, 0, 3);              // -> global_prefetch_b8
    int node = 0;
#pragma unroll
    for (int d = 0; d < MAX_DEPTH; ++d) {
      const int   fi  = featL[node];
      const float thr = thrL[node];
      const float v   = xr[fi];                      // near-cache hit
      node = 2 * node + 1 + (v > thr ? 1 : 0);
    }
    leaf = node - N_NODES;
  }
  leafL[tid] = leaf;
  __syncthreads();

  // ---- Phase 2: B128 coalesced row copy from L2-resident softmax table -----
  int valid = batch - s0;
  if (valid > BLK) valid = BLK;
  const int total4 = valid * N_CLS4;                 // float4 units
  const vf4* tbl4  = (const vf4*)table;              // rows are 16B-aligned
  vf4* orow4 = (vf4*)(out + (size_t)s0 * N_CLASSES); // block slab 16B-aligned
  for (int i = tid; i < total4; i += BLK) {
    const unsigned s = (unsigned)i / (unsigned)N_CLS4;  // magic-mul div
    const int      j = i - (int)s * N_CLS4;
    const vf4      v = tbl4[(size_t)leafL[s] * N_CLS4 + j];
    __builtin_nontemporal_store(v, orow4 + i);       // streaming 200MB output
  }
}

// ---------------------------------------------------------------------------
extern "C" void kernel_launch(void* const* d_in, const int* in_sizes, int n_in,
                              void* d_out, int out_size, void* d_ws, size_t ws_size,
                              hipStream_t stream) {
  const float* x  = (const float*)d_in[0];  // [B, 256]
  const float* sf = (const float*)d_in[1];  // [4095]
  const float* st = (const float*)d_in[2];  // [4095]
  const float* lp = (const float*)d_in[3];  // [4096, 100]
  float* out = (float*)d_out;

  const int batch = in_sizes[0] / INPUT_DIM;
  float* table = (float*)d_ws;              // 4096*100*4 = 1.6 MB scratch

  // Kernel 1: 4096 rows, 8 waves/block -> 512 blocks.
  dt_softmax_table_kernel<<<N_LEAVES / (BLK >> 5), BLK, 0, stream>>>(lp, table);

  // Kernel 2: one 256-thread block per 256 samples.
  const int nblk = (batch + BLK - 1) / BLK;
  dt_traverse_kernel<<<nblk, BLK, 0, stream>>>(x, sf, st, table, out, batch);

  (void)n_in; (void)out_size; (void)ws_size;
}